// VMamba3DBlock_75995151335666
// MI455X (gfx1250) — compile-verified
//
#include <hip/hip_runtime.h>

static constexpr int    S     = 64;
static constexpr int    C     = 64;
static constexpr int    NSPAT = S * S * S;              // 262144 spatial sites
static constexpr size_t NCSZ  = (size_t)NSPAT * C;      // 16777216 elements
static constexpr float  EPSN  = 1e-5f;

typedef __attribute__((ext_vector_type(2))) float v2f;
typedef __attribute__((ext_vector_type(8))) float v8f;

__device__ __forceinline__ float dev_sigmoid(float x) { return 1.0f / (1.0f + expf(-x)); }
__device__ __forceinline__ float dev_softplus(float x) {
  return fmaxf(x, 0.0f) + log1pf(expf(-fabsf(x)));
}

// ---------------- init: zero atomic slots, precompute softplus(A) ----------------
__global__ void k_init(const float* __restrict__ A, float* __restrict__ stats,
                       float* __restrict__ Apos) {
  int t = threadIdx.x;
  if (t < 8) stats[t] = 0.0f;
  if (t < C) Apos[t] = dev_softplus(A[t]);
}

// ---------------- generic global sum / sumsq reduction ----------------
__global__ __launch_bounds__(256) void k_reduce(const float* __restrict__ src,
                                                float* __restrict__ stats, int off, size_t n) {
  __shared__ float ssum[256];
  __shared__ float ssq[256];
  size_t stride = (size_t)gridDim.x * blockDim.x;
  float ls = 0.0f, lq = 0.0f;
  for (size_t i = (size_t)blockIdx.x * blockDim.x + threadIdx.x; i < n; i += stride) {
    float v = src[i];
    ls += v;
    lq += v * v;
  }
  ssum[threadIdx.x] = ls;
  ssq[threadIdx.x]  = lq;
  __syncthreads();
  for (int w = 128; w > 0; w >>= 1) {
    if (threadIdx.x < (unsigned)w) {
      ssum[threadIdx.x] += ssum[threadIdx.x + w];
      ssq[threadIdx.x]  += ssq[threadIdx.x + w];
    }
    __syncthreads();
  }
  if (threadIdx.x == 0) {
    atomicAdd(&stats[off],     ssum[0]);
    atomicAdd(&stats[off + 1], ssq[0]);
  }
}

__global__ void k_finalize(float* __restrict__ stats, int off) {
  float invM = 1.0f / (float)NCSZ;
  float mu   = stats[off] * invM;
  float var  = stats[off + 1] * invM - mu * mu;
  stats[off + 2] = mu;
  stats[off + 3] = rsqrtf(var + EPSN);
}

// ---------------- [C,N] -> [N,C] transpose (also zero-inits y) ----------------
__global__ __launch_bounds__(256) void k_transpose_in(const float* __restrict__ x,
                                                      float* __restrict__ xt,
                                                      float* __restrict__ ybuf) {
  __shared__ float sm[32][33];
  int n0 = blockIdx.x * 32, c0 = blockIdx.y * 32;
  int tx = threadIdx.x, ty = threadIdx.y;
  for (int i = ty; i < 32; i += 8)
    sm[i][tx] = x[(size_t)(c0 + i) * NSPAT + (n0 + tx)];   // coalesced along n
  __syncthreads();
  for (int i = ty; i < 32; i += 8) {
    size_t idx = (size_t)(n0 + i) * C + (c0 + tx);         // coalesced along c
    xt[idx]   = sm[tx][i];
    ybuf[idx] = 0.0f;
  }
}

// ---------------- proj: z = GN1(x); params = z @ pw^T + pb; split+activations ----------------
// 16-row tile per block; 5 waves x 4 column-tiles = 320 output channels; K = 64 (16 WMMA steps)
__global__ __launch_bounds__(160) void k_proj(
    const float* __restrict__ xt, const float* __restrict__ stats,
    const float* __restrict__ n1w, const float* __restrict__ n1b,
    const float* __restrict__ pw, const float* __restrict__ pb,
    const float* __restrict__ Apos,
    float* __restrict__ ubuf, float* __restrict__ gout,
    float* __restrict__ Bbuf, float* __restrict__ Cbuf, float* __restrict__ albuf) {
  __shared__ float zt[16 * 65];   // padded stride 65: conflict-free A-operand reads
  int tile = blockIdx.x;
  int tid  = threadIdx.x;
  float mu = stats[2], rs = stats[3];
  for (int i = tid; i < 16 * C; i += 160) {
    int m = i >> 6, c = i & 63;
    float v = xt[(size_t)tile * (16 * C) + i];
    zt[m * 65 + c] = (v - mu) * rs * n1w[c] + n1b[c];
  }
  __syncthreads();
  int wave = tid >> 5, lane = tid & 31;
  int m = lane & 15, g = lane >> 4;
  v8f vzero = {0.f, 0.f, 0.f, 0.f, 0.f, 0.f, 0.f, 0.f};
  v8f acc[4];
  for (int t = 0; t < 4; ++t) acc[t] = vzero;
  for (int k = 0; k < 16; ++k) {
    int c0 = 4 * k + 2 * g;
    v2f a;
    a.x = zt[m * 65 + c0];
    a.y = zt[m * 65 + c0 + 1];
#pragma unroll
    for (int t = 0; t < 4; ++t) {
      int o = (wave * 4 + t) * 16 + m;
      v2f b;
      b.x = pw[o * C + c0];
      b.y = pw[o * C + c0 + 1];
      acc[t] = __builtin_amdgcn_wmma_f32_16x16x4_f32(false, a, false, b, (short)0,
                                                     acc[t], false, false);
    }
  }
#pragma unroll
  for (int t = 0; t < 4; ++t) {
    int o    = (wave * 4 + t) * 16 + m;
    int sect = o >> 6;       // 0:g_in 1:g_out 2:B 3:C 4:Delta  (tile-uniform)
    int c    = o & 63;
    float bias = pb[o];
#pragma unroll
    for (int r = 0; r < 8; ++r) {
      int mrow  = r + 8 * g;
      float p   = acc[t][r] + bias;
      size_t idx = ((size_t)tile * 16 + mrow) * C + c;
      if (sect == 0)      ubuf[idx] = dev_sigmoid(p) * zt[mrow * 65 + c];
      else if (sect == 1) gout[idx] = dev_sigmoid(p);
      else if (sect == 2) Bbuf[idx] = p;
      else if (sect == 3) Cbuf[idx] = p;
      else {
        float delta = dev_softplus(p) + 1e-4f;
        albuf[idx]  = expf(-Apos[c] * delta);
      }
    }
  }
}

// ---------------- bidirectional line scan along one axis ----------------
// 64 threads = 64 channels of one line; operands cached in LDS for the backward pass.
// forward adds 0.5*C*s_f + D*u ; backward adds 0.5*C*s_b  (y = C*(s_f+s_b)/2 + D*u)
__global__ __launch_bounds__(64) void k_scan(
    const float* __restrict__ ubuf, const float* __restrict__ Bbuf,
    const float* __restrict__ Cbuf, const float* __restrict__ albuf,
    const float* __restrict__ Dp, float* __restrict__ ybuf,
    int ddiv, int m1, int m2, int stepN) {
  __shared__ float sA[64 * 64];
  __shared__ float sBU[64 * 64];
  __shared__ float sC[64 * 64];
  int L = blockIdx.x;
  int c = threadIdx.x;
  int base = (L / ddiv) * m1 + (L % ddiv) * m2;
  float Dc = Dp[c];
  float s = 0.0f;
  for (int t = 0; t < 64; ++t) {
    size_t idx = ((size_t)(base + t * stepN)) * C + c;
    float a = albuf[idx], b = Bbuf[idx], uu = ubuf[idx], cc = Cbuf[idx];
    float bu = b * uu;
    sA[t * 64 + c]  = a;
    sBU[t * 64 + c] = bu;
    sC[t * 64 + c]  = cc;
    s = a * s + bu;
    ybuf[idx] += 0.5f * cc * s + Dc * uu;
  }
  s = 0.0f;
  for (int t = 63; t >= 0; --t) {
    float a = sA[t * 64 + c], bu = sBU[t * 64 + c], cc = sC[t * 64 + c];
    s = a * s + bu;
    size_t idx = ((size_t)(base + t * stepN)) * C + c;
    ybuf[idx] += 0.5f * cc * s;
  }
}

// ---------------- xc2 = xt + g_out*y/3 ; fused GroupNorm2 reduction ----------------
__global__ __launch_bounds__(256) void k_xc2(const float* __restrict__ xt,
                                             const float* __restrict__ gout,
                                             const float* __restrict__ ybuf,
                                             float* __restrict__ xc2,
                                             float* __restrict__ stats) {
  __shared__ float ssum[256];
  __shared__ float ssq[256];
  size_t stride = (size_t)gridDim.x * blockDim.x;
  float ls = 0.0f, lq = 0.0f;
  for (size_t i = (size_t)blockIdx.x * blockDim.x + threadIdx.x; i < NCSZ; i += stride) {
    float v = xt[i] + gout[i] * ybuf[i] * (1.0f / 3.0f);
    xc2[i] = v;
    ls += v;
    lq += v * v;
  }
  ssum[threadIdx.x] = ls;
  ssq[threadIdx.x]  = lq;
  __syncthreads();
  for (int w = 128; w > 0; w >>= 1) {
    if (threadIdx.x < (unsigned)w) {
      ssum[threadIdx.x] += ssum[threadIdx.x + w];
      ssq[threadIdx.x]  += ssq[threadIdx.x + w];
    }
    __syncthreads();
  }
  if (threadIdx.x == 0) {
    atomicAdd(&stats[4], ssum[0]);
    atomicAdd(&stats[5], ssq[0]);
  }
}

// ---------------- MLP: GN2 -> W1 -> exact GELU -> W2 -> residual ----------------
// 16-row tile per block; 4 waves. matmul1: 16 col-tiles (4/wave, K=64).
// matmul2: 4 col-tiles (1/wave, K=256) with A from padded LDS.
__global__ __launch_bounds__(128) void k_mlp(
    const float* __restrict__ xc2, const float* __restrict__ stats,
    const float* __restrict__ n2w, const float* __restrict__ n2b,
    const float* __restrict__ w1, const float* __restrict__ b1,
    const float* __restrict__ w2, const float* __restrict__ b2,
    float* __restrict__ outc) {
  __shared__ float z2[16 * 65];    // padded
  __shared__ float hb[16 * 257];   // padded
  int tile = blockIdx.x;
  int tid  = threadIdx.x;
  float mu = stats[6], rs = stats[7];
  for (int i = tid; i < 16 * C; i += 128) {
    int m = i >> 6, c = i & 63;
    float v = xc2[(size_t)tile * (16 * C) + i];
    z2[m * 65 + c] = (v - mu) * rs * n2w[c] + n2b[c];
  }
  __syncthreads();
  int wave = tid >> 5, lane = tid & 31;
  int m = lane & 15, g = lane >> 4;
  v8f vzero = {0.f, 0.f, 0.f, 0.f, 0.f, 0.f, 0.f, 0.f};
  v8f acc[4];
  for (int t = 0; t < 4; ++t) acc[t] = vzero;
  for (int k = 0; k < 16; ++k) {
    int c0 = 4 * k + 2 * g;
    v2f a;
    a.x = z2[m * 65 + c0];
    a.y = z2[m * 65 + c0 + 1];
#pragma unroll
    for (int t = 0; t < 4; ++t) {
      int o = (wave * 4 + t) * 16 + m;
      v2f b;
      b.x = w1[o * C + c0];
      b.y = w1[o * C + c0 + 1];
      acc[t] = __builtin_amdgcn_wmma_f32_16x16x4_f32(false, a, false, b, (short)0,
                                                     acc[t], false, false);
    }
  }
#pragma unroll
  for (int t = 0; t < 4; ++t) {
    int o = (wave * 4 + t) * 16 + m;
    float bias = b1[o];
#pragma unroll
    for (int r = 0; r < 8; ++r) {
      int mrow = r + 8 * g;
      float h  = acc[t][r] + bias;
      h = 0.5f * h * (1.0f + erff(h * 0.70710678118654752f));  // exact GELU
      hb[mrow * 257 + o] = h;
    }
  }
  __syncthreads();
  v8f acc2 = vzero;
  int cout = wave * 16 + m;
  for (int k = 0; k < 64; ++k) {
    int c0 = 4 * k + 2 * g;
    v2f a;
    a.x = hb[m * 257 + c0];
    a.y = hb[m * 257 + c0 + 1];
    v2f b;
    b.x = w2[cout * 256 + c0];
    b.y = w2[cout * 256 + c0 + 1];
    acc2 = __builtin_amdgcn_wmma_f32_16x16x4_f32(false, a, false, b, (short)0,
                                                 acc2, false, false);
  }
  float bias2 = b2[cout];
#pragma unroll
  for (int r = 0; r < 8; ++r) {
    int mrow  = r + 8 * g;
    size_t idx = ((size_t)tile * 16 + mrow) * C + cout;
    outc[idx] = acc2[r] + bias2 + xc2[idx];
  }
}

// ---------------- [N,C] -> [C,N] output transpose ----------------
__global__ __launch_bounds__(256) void k_transpose_out(const float* __restrict__ outc,
                                                       float* __restrict__ out) {
  __shared__ float sm[32][33];
  int n0 = blockIdx.x * 32, c0 = blockIdx.y * 32;
  int tx = threadIdx.x, ty = threadIdx.y;
  for (int i = ty; i < 32; i += 8)
    sm[i][tx] = outc[(size_t)(n0 + i) * C + (c0 + tx)];    // coalesced along c
  __syncthreads();
  for (int i = ty; i < 32; i += 8)
    out[(size_t)(c0 + i) * NSPAT + (n0 + tx)] = sm[tx][i]; // coalesced along n
}

extern "C" void kernel_launch(void* const* d_in, const int* in_sizes, int n_in,
                              void* d_out, int out_size, void* d_ws, size_t ws_size,
                              hipStream_t stream) {
  (void)in_sizes; (void)n_in; (void)out_size; (void)ws_size;
  const float* x   = (const float*)d_in[0];
  const float* n1w = (const float*)d_in[1];
  const float* n1b = (const float*)d_in[2];
  const float* pw  = (const float*)d_in[3];
  const float* pb  = (const float*)d_in[4];
  const float* A   = (const float*)d_in[5];
  const float* Dp  = (const float*)d_in[6];
  const float* n2w = (const float*)d_in[7];
  const float* n2b = (const float*)d_in[8];
  const float* w1  = (const float*)d_in[9];
  const float* b1  = (const float*)d_in[10];
  const float* w2  = (const float*)d_in[11];
  const float* b2  = (const float*)d_in[12];
  float* out = (float*)d_out;

  float* ws    = (float*)d_ws;
  float* stats = ws;          // 8 floats: sum1,sq1,mu1,rs1,sum2,sq2,mu2,rs2
  float* Apos  = ws + 64;     // 64 floats
  float* base  = ws + 256;
  float* xt    = base;                // [N,C]
  float* ubuf  = xt    + NCSZ;
  float* gout  = ubuf  + NCSZ;
  float* Bbuf  = gout  + NCSZ;
  float* Cbuf  = Bbuf  + NCSZ;
  float* albuf = Cbuf  + NCSZ;
  float* ybuf  = albuf + NCSZ;
  float* xc2   = ybuf  + NCSZ;
  float* outc  = xc2   + NCSZ;

  k_init<<<1, 64, 0, stream>>>(A, stats, Apos);
  k_reduce<<<4096, 256, 0, stream>>>(x, stats, 0, NCSZ);
  k_finalize<<<1, 1, 0, stream>>>(stats, 0);

  dim3 tb(32, 8);
  k_transpose_in<<<dim3(NSPAT / 32, 2), tb, 0, stream>>>(x, xt, ybuf);

  k_proj<<<NSPAT / 16, 160, 0, stream>>>(xt, stats, n1w, n1b, pw, pb, Apos,
                                         ubuf, gout, Bbuf, Cbuf, albuf);

  // n = d*4096 + h*64 + w ; line base = (L/ddiv)*m1 + (L%ddiv)*m2, element step = stepN
  k_scan<<<4096, 64, 0, stream>>>(ubuf, Bbuf, Cbuf, albuf, Dp, ybuf, 4096, 0, 64, 1);   // W axis
  k_scan<<<4096, 64, 0, stream>>>(ubuf, Bbuf, Cbuf, albuf, Dp, ybuf, 64, 4096, 1, 64);  // H axis
  k_scan<<<4096, 64, 0, stream>>>(ubuf, Bbuf, Cbuf, albuf, Dp, ybuf, 4096, 0, 1, 4096); // D axis

  k_xc2<<<4096, 256, 0, stream>>>(xt, gout, ybuf, xc2, stats);
  k_finalize<<<1, 1, 0, stream>>>(stats, 4);

  k_mlp<<<NSPAT / 16, 128, 0, stream>>>(xc2, stats, n2w, n2b, w1, b1, w2, b2, outc);

  k_transpose_out<<<dim3(NSPAT / 32, 2), tb, 0, stream>>>(outc, out);
}